// Yolov4Layer_33466385170571
// MI455X (gfx1250) — compile-verified
//
#include <hip/hip_runtime.h>
#include <math.h>

#define NAA 18
#define NCC 80
#define GSZ 76
#define GG  (GSZ * GSZ)        // 5776
#define NCH (NCC + 6)          // 86
#define PI_F 3.14159265358979323846f

typedef __attribute__((ext_vector_type(2))) float v2f;
typedef __attribute__((ext_vector_type(8))) float v8f;

// MASKED_ANCHORS: (w/8, h/8) per anchor x 6 angles, anchor-major
__device__ __constant__ float d_aw[NAA] = {
    1.5f, 1.5f, 1.5f, 1.5f, 1.5f, 1.5f,
    2.375f, 2.375f, 2.375f, 2.375f, 2.375f, 2.375f,
    5.0f, 5.0f, 5.0f, 5.0f, 5.0f, 5.0f};
__device__ __constant__ float d_ah[NAA] = {
    2.0f, 2.0f, 2.0f, 2.0f, 2.0f, 2.0f,
    4.5f, 4.5f, 4.5f, 4.5f, 4.5f, 4.5f,
    3.5f, 3.5f, 3.5f, 3.5f, 3.5f, 3.5f};
__device__ __constant__ float d_aa[NAA] = {
    -PI_F/3.f, -PI_F/6.f, 0.f, PI_F/6.f, PI_F/3.f, PI_F/2.f,
    -PI_F/3.f, -PI_F/6.f, 0.f, PI_F/6.f, PI_F/3.f, PI_F/2.f,
    -PI_F/3.f, -PI_F/6.f, 0.f, PI_F/6.f, PI_F/3.f, PI_F/2.f};

__device__ __forceinline__ float sigmoidf_(float x) { return 1.0f / (1.0f + expf(-x)); }
__device__ __forceinline__ float clip01(float p) {
    return fminf(fmaxf(p, 1e-7f), 1.0f - 1e-7f);
}

// ---- wave32 sum via V_WMMA_F32_16X16X4_F32: A=ones(16x4), B=values(4x16).
// D[m][n] = colsum[n] = v[lane n] + v[lane n+16]; 4 shfl_xor steps finish 16->1.
// EXEC must be all ones at the call site.
__device__ __forceinline__ float wave_sum32_wmma(float v) {
    v2f a; a[0] = 1.0f; a[1] = 1.0f;
    v2f b; b[0] = v;    b[1] = 0.0f;
    v8f c = {0.f,0.f,0.f,0.f,0.f,0.f,0.f,0.f};
    c = __builtin_amdgcn_wmma_f32_16x16x4_f32(false, a, false, b, (short)0, c, false, false);
    float s = c[0];
    s += __shfl_xor(s, 8, 32);
    s += __shfl_xor(s, 4, 32);
    s += __shfl_xor(s, 2, 32);
    s += __shfl_xor(s, 1, 32);
    return s;  // total over 32 lanes, valid in all lanes
}

__device__ __forceinline__ int cell_idx(int b, int a, int j, int i) {
    return ((b * NAA + a) * GSZ + j) * GSZ + i;
}

// ---------------- K0: init workspace ----------------
__global__ void k_init(int* __restrict__ winner, float* __restrict__ noobj,
                       float* __restrict__ acc, int ncells) {
    int idx = blockIdx.x * blockDim.x + threadIdx.x;
    if (idx < 8) acc[idx] = 0.0f;
    if (idx < ncells) {
        winner[idx] = -1;
        noobj[idx]  = 1.0f;
    }
}

// ---------------- K1: anchor matching / mask scatter ----------------
__global__ void k_match(const float* __restrict__ tgt, int* __restrict__ winner,
                        float* __restrict__ noobj, int nT) {
    int t = blockIdx.x * blockDim.x + threadIdx.x;
    if (t >= nT) return;
    const float* tr = tgt + (size_t)t * 7;
    int   b  = (int)tr[0];
    float gx = tr[2] * (float)GSZ, gy = tr[3] * (float)GSZ;
    float gw = tr[4] * (float)GSZ, gh = tr[5] * (float)GSZ;
    float ga = tr[6];
    int gi = (int)gx; gi = min(max(gi, 0), GSZ - 1);
    int gj = (int)gy; gj = min(max(gj, 0), GSZ - 1);

    float best = -1e30f; int best_n = 0;
    for (int a = 0; a < NAA; ++a) {
        float aw = d_aw[a], ah = d_ah[a], aa = d_aa[a];
        float inter = fminf(aw, gw) * fminf(ah, gh);
        float uni   = aw * ah + 1e-16f + gw * gh - inter;
        float off   = fabsf(aa - ga);
        float ar    = inter / uni * fabsf(cosf(aa - ga));
        if (ar > best) { best = ar; best_n = a; }
        bool cond = (ar > 0.5f) || ((ar > 0.4f) && (off < PI_F / 12.0f));
        if (cond) noobj[cell_idx(b, a, gj, gi)] = 0.0f;   // only lowers to 0: race-safe
    }
    int ci = cell_idx(b, best_n, gj, gi);
    noobj[ci] = 0.0f;
    atomicMax(&winner[ci], t);    // "last target wins" scatter semantics
}

// ---------------- K3: streaming decode + focal-conf reduction ----------------
__global__ __launch_bounds__(256)
void k_decode(const float* __restrict__ src, const int* __restrict__ winner,
              const float* __restrict__ noobj, float* __restrict__ dec,
              float* __restrict__ acc, int ncells) {
    int cell = blockIdx.x * blockDim.x + threadIdx.x;
    bool active = cell < ncells;
    int cc = active ? cell : 0;

    int i = cc % GSZ;
    int j = (cc / GSZ) % GSZ;
    int a = (cc / GG) % NAA;
    int b = cc / (GG * NAA);

    size_t srcbase = ((size_t)(b * NAA + a) * NCH) * (size_t)GG + (size_t)j * GSZ + i;

    float x0 = src[srcbase];
    float x1 = src[srcbase + (size_t)GG];
    float x2 = src[srcbase + 2 * (size_t)GG];
    float x3 = src[srcbase + 3 * (size_t)GG];
    float x4 = src[srcbase + 4 * (size_t)GG];
    float x5 = src[srcbase + 5 * (size_t)GG];
    __builtin_prefetch(&src[srcbase + 6 * (size_t)GG], 0, 1);  // global_prefetch_b8

    float px = sigmoidf_(x0) * 1.05f - 0.025f + (float)i;
    float py = sigmoidf_(x1) * 1.05f - 0.025f + (float)j;
    float pw = expf(x2) * d_aw[a];
    float ph = expf(x3) * d_ah[a];
    float pa = x4 + d_aa[a];
    float pconf = sigmoidf_(x5);

    if (active) {
        // 86 floats per cell, base 8B-aligned (344B stride): pack as 43x b64 NT stores
        v2f* dp2 = (v2f*)(dec + (size_t)cell * NCH);
        v2f w0; w0[0] = px * 8.0f; w0[1] = py * 8.0f;
        v2f w1; w1[0] = pw * 8.0f; w1[1] = ph * 8.0f;
        v2f w2; w2[0] = pa;        w2[1] = pconf;
        __builtin_nontemporal_store(w0, dp2 + 0);
        __builtin_nontemporal_store(w1, dp2 + 1);
        __builtin_nontemporal_store(w2, dp2 + 2);
        #pragma unroll 8
        for (int c = 0; c < NCC; c += 2) {
            float a0 = __builtin_nontemporal_load(&src[srcbase + (size_t)(6 + c) * GG]);
            float a1 = __builtin_nontemporal_load(&src[srcbase + (size_t)(7 + c) * GG]);
            v2f wv; wv[0] = sigmoidf_(a0); wv[1] = sigmoidf_(a1);
            __builtin_nontemporal_store(wv, dp2 + 3 + (c >> 1));
        }
    }

    // focal conf loss terms
    float fl_obj = 0.0f, fl_no = 0.0f, objf = 0.0f, nov = 0.0f;
    if (active) {
        float p = clip01(pconf);
        int   w = winner[cell];
        nov = noobj[cell];
        if (w >= 0) {
            objf = 1.0f;
            float om = 1.0f - p;
            fl_obj = -logf(p) * 0.25f * om * om;     // t=1 focal BCE
            nov = 0.0f;                               // noobj forced 0 at obj cells
        } else if (nov > 0.0f) {
            fl_no = -logf(1.0f - p) * 0.75f * p * p; // t=0 focal BCE
        }
    }

    // EXEC all-ones here (divergence closed) -> WMMA reductions legal.
    // Four back-to-back identical wave-level WMMA sums, then one LDS pass.
    float s0 = wave_sum32_wmma(fl_obj);
    float s1 = wave_sum32_wmma(fl_no);
    float s2 = wave_sum32_wmma(objf);
    float s3 = wave_sum32_wmma(nov);

    __shared__ float smacc[4];
    if (threadIdx.x < 4) smacc[threadIdx.x] = 0.0f;
    __syncthreads();
    if ((threadIdx.x & 31) == 0) {
        atomicAdd(&smacc[0], s0);   // ds_add_f32
        atomicAdd(&smacc[1], s1);
        atomicAdd(&smacc[2], s2);
        atomicAdd(&smacc[3], s3);
    }
    __syncthreads();
    if (threadIdx.x == 0) {
        atomicAdd(&acc[0], smacc[0]);
        atomicAdd(&acc[1], smacc[1]);
        atomicAdd(&acc[2], smacc[2]);
        atomicAdd(&acc[3], smacc[3]);
    }
}

// ---------------- K4: per-target reg(skew) + cls loss ----------------
__global__ void k_target_loss(const float* __restrict__ src, const float* __restrict__ tgt,
                              const int* __restrict__ winner, float* __restrict__ acc) {
    int t = blockIdx.x;
    const float* tr = tgt + (size_t)t * 7;
    int   b  = (int)tr[0];
    int   lab = (int)tr[1];
    float gx = tr[2] * (float)GSZ, gy = tr[3] * (float)GSZ;
    float gw = tr[4] * (float)GSZ, gh = tr[5] * (float)GSZ;
    float ga = tr[6];
    int gi = (int)gx; gi = min(max(gi, 0), GSZ - 1);
    int gj = (int)gy; gj = min(max(gj, 0), GSZ - 1);

    // recompute best anchor (cheap, uniform across block)
    float best = -1e30f; int best_n = 0;
    for (int a = 0; a < NAA; ++a) {
        float inter = fminf(d_aw[a], gw) * fminf(d_ah[a], gh);
        float uni   = d_aw[a] * d_ah[a] + 1e-16f + gw * gh - inter;
        float ar    = inter / uni * fabsf(cosf(d_aa[a] - ga));
        if (ar > best) { best = ar; best_n = a; }
    }
    int ci = cell_idx(b, best_n, gj, gi);
    if (winner[ci] != t) return;   // uniform: a duplicate-scattered cell, later target won

    size_t srcbase = ((size_t)(b * NAA + best_n) * NCH) * (size_t)GG + (size_t)gj * GSZ + gi;

    int tid = threadIdx.x;
    __shared__ float sm[128];
    float clsv = 0.0f;
    if (tid < NCC) {
        float p = clip01(sigmoidf_(src[srcbase + (size_t)(6 + tid) * GG]));
        clsv = (tid == lab) ? -logf(p) : -logf(1.0f - p);  // BCE vs one-hot
    }
    sm[tid] = clsv;
    __syncthreads();
    for (int s = 64; s > 0; s >>= 1) {
        if (tid < s) sm[tid] += sm[tid + s];
        __syncthreads();
    }
    if (tid == 0) atomicAdd(&acc[5], sm[0]);

    if (tid == 127) {
        // reg loss contribution collapses to skew = exp(1 - skew_iou) - 1
        float x0 = src[srcbase];
        float x1 = src[srcbase + (size_t)GG];
        float x2 = src[srcbase + 2 * (size_t)GG];
        float x3 = src[srcbase + 3 * (size_t)GG];
        float x4 = src[srcbase + 4 * (size_t)GG];
        float px = sigmoidf_(x0) * 1.05f - 0.025f + (float)gi;
        float py = sigmoidf_(x1) * 1.05f - 0.025f + (float)gj;
        float pw = expf(x2) * d_aw[best_n];
        float ph = expf(x3) * d_ah[best_n];
        float pa = x4 + d_aa[best_n];
        // axis-aligned IoU (grid units)
        float p1x = px - pw * 0.5f, p1y = py - ph * 0.5f;
        float p2x = px + pw * 0.5f, p2y = py + ph * 0.5f;
        float t1x = gx - gw * 0.5f, t1y = gy - gh * 0.5f;
        float t2x = gx + gw * 0.5f, t2y = gy + gh * 0.5f;
        float ix = fmaxf(fminf(p2x, t2x) - fmaxf(p1x, t1x), 0.0f);
        float iy = fmaxf(fminf(p2y, t2y) - fmaxf(p1y, t1y), 0.0f);
        float inter = ix * iy;
        float uni = pw * ph + gw * gh - inter;
        float iou = inter / uni;
        float skew_iou = iou * fabsf(cosf(pa - ga));
        atomicAdd(&acc[4], expf(1.0f - skew_iou) - 1.0f);
    }
}

// ---------------- K5: finalize scalar loss ----------------
__global__ void k_final(const float* __restrict__ acc, float* __restrict__ dec, int ncells) {
    if (threadIdx.x != 0 || blockIdx.x != 0) return;
    float n_obj   = fmaxf(acc[2], 1.0f);
    float n_noobj = fmaxf(acc[3], 1.0f);
    float conf = acc[0] / n_obj + acc[1] / n_noobj;
    float reg  = acc[4] / n_obj;
    float cls  = acc[5] / (n_obj * (float)NCC);
    dec[(size_t)ncells * NCH] = 1.0f * reg + 10.0f * conf + 1.0f * cls;
}

extern "C" void kernel_launch(void* const* d_in, const int* in_sizes, int n_in,
                              void* d_out, int out_size, void* d_ws, size_t ws_size,
                              hipStream_t stream) {
    const float* src = (const float*)d_in[0];
    const float* tgt = (const float*)d_in[1];
    float* dec = (float*)d_out;

    int Bv = in_sizes[0] / (NAA * NCH * GG);   // 8
    int nT = in_sizes[1] / 7;                  // 240
    int ncells = Bv * NAA * GG;                // 831744

    float* acc    = (float*)d_ws;
    int*   winner = (int*)((char*)d_ws + 64);
    float* noobj  = (float*)((char*)d_ws + 64 + (size_t)ncells * sizeof(int));

    int tb = 256, nb = (ncells + tb - 1) / tb;
    k_init<<<nb, tb, 0, stream>>>(winner, noobj, acc, ncells);
    k_match<<<(nT + 63) / 64, 64, 0, stream>>>(tgt, winner, noobj, nT);
    k_decode<<<nb, tb, 0, stream>>>(src, winner, noobj, dec, acc, ncells);
    k_target_loss<<<nT, 128, 0, stream>>>(src, tgt, winner, acc);
    k_final<<<1, 1, 0, stream>>>(acc, dec, ncells);
}